// CRFLoss_3753801417182
// MI455X (gfx1250) — compile-verified
//
#include <hip/hip_runtime.h>
#include <hip/hip_bf16.h>
#include <cstdint>
#include <cstddef>

#define LL 43
#define CC 46
#define BB 128
#define TT 4096
#define SEG 32          // segments per batch
#define SEGLEN 128      // timesteps per segment (SEG*SEGLEN == TT)
#define NEGF (-1.0e30f)

typedef float v2f __attribute__((ext_vector_type(2)));
typedef float v8f __attribute__((ext_vector_type(8)));

// Workspace (floats):
// [0..47]   w0 = exp(s0_lab) padded to 48
// [48..95]  w1 = exp(s1_lab) padded to 48
// [96]=sO  [97]=s0I  [98]=s1I  [99]=sfin
// [128 ..)                      segment matrices  (B*SEG*4 floats)
// [128+B*SEG*4 ..)              per-block num partials (B*SEG floats)
#define WS_SEGMAT 128
#define WS_NUMP   (WS_SEGMAT + BB * SEG * 4)

__device__ __forceinline__ float lse2(float x, float y) {
    float mx = fmaxf(x, y);
    float mn = fminf(x, y);
    return mx + log1pf(__expf(mn - mx));
}

__global__ void crf_consts_kernel(const float* __restrict__ p, float* __restrict__ cw) {
    if (threadIdx.x == 0) {
        // g0 = log_softmax(p[0:46])
        float m0 = -3.4e38f;
        for (int i = 0; i < LL + 3; ++i) m0 = fmaxf(m0, p[i]);
        float s = 0.f;
        for (int i = 0; i < LL + 3; ++i) s += __expf(p[i] - m0);
        float lse0 = m0 + __logf(s);
        cw[96] = p[0] - lse0;          // sO
        cw[97] = p[LL + 1] - lse0;     // s0I
        cw[99] = p[LL + 2] - lse0;     // sfin
        for (int l = 0; l < 48; ++l)
            cw[l] = (l < LL) ? __expf(p[1 + l] - lse0) : 0.f;
        // g1 = log_softmax(p[46:90])
        const float* q = p + LL + 3;
        float m1 = -3.4e38f;
        for (int i = 0; i < LL + 1; ++i) m1 = fmaxf(m1, q[i]);
        float s1 = 0.f;
        for (int i = 0; i < LL + 1; ++i) s1 += __expf(q[i] - m1);
        float lse1 = m1 + __logf(s1);
        cw[98] = q[0] - lse1;          // s1I
        for (int l = 0; l < 48; ++l)
            cw[48 + l] = (l < LL) ? __expf(q[1 + l] - lse1) : 0.f;
    }
}

__global__ __launch_bounds__(256) void crf_seg_kernel(
    const float* __restrict__ lp, const int* __restrict__ lens,
    const int* __restrict__ labels, const float* __restrict__ cw,
    float* __restrict__ segMat, float* __restrict__ numPart)
{
    __shared__ __align__(16) float ldsE[128][50];   // exp(lp_lab), K padded to 48 (+2 anti-bank-conflict)
    __shared__ __align__(16) float ldsLp[128][2];   // lp[:,1], lp[:,2]
    __shared__ __align__(16) float ldsW[96];        // w0[48], w1[48]
    __shared__ __align__(16) float ldsDot[8][32];   // per-wave: [col*16+row]
    __shared__ __align__(16) float ldsM[128][4];    // 2x2 log-semiring matrices per t
    __shared__ __align__(16) float ldsNum[256];

    const int tid   = threadIdx.x;
    const int lane  = tid & 31;
    const int w     = tid >> 5;
    const int m     = lane & 15;
    const int khalf = lane >> 4;

    const int bid = blockIdx.x;
    const int b   = bid / SEG;
    const int seg = bid - b * SEG;
    const int t0  = seg * SEGLEN;

    if (tid < 96) ldsW[tid] = cw[tid];
    if (tid < 128) {
        #pragma unroll
        for (int k = LL; k < 48; ++k) ldsE[tid][k] = 0.f;
    }

    // Stream 16 rows x 46 channels = 736 contiguous floats per wave (16B-aligned).
    const float*  rowBase  = lp + ((size_t)b * TT + (size_t)(t0 + w * 16)) * CC;
    const float4* rowBase4 = (const float4*)rowBase;
    #pragma unroll
    for (int it = 0; it < 6; ++it) {
        int idx = lane + 32 * it;
        if (idx < 184) {
            float4 v4 = rowBase4[idx];
            float vv[4] = {v4.x, v4.y, v4.z, v4.w};
            #pragma unroll
            for (int e = 0; e < 4; ++e) {
                int g   = idx * 4 + e;
                int row = g / CC;
                int c   = g - row * CC;
                float v = vv[e];
                if (c >= 3)       ldsE[w * 16 + row][c - 3] = __expf(v);
                else if (c == 1)  ldsLp[w * 16 + row][0] = v;
                else if (c == 2)  ldsLp[w * 16 + row][1] = v;
            }
        }
    }

    // numerator contribution: lanes 0..15 of each wave own one row each
    const int myT = t0 + w * 16 + m;
    float tok = 0.f;
    if (lane < 16) {
        int   lab = labels[(size_t)b * TT + myT];
        float tv  = lp[((size_t)b * TT + myT) * CC + lab];
        tok = (myT < lens[b]) ? tv : 0.f;
    }

    __syncthreads();

    // WMMA: D[16x16] = E[16x48] x W[48x16], W col0=w0, col1=w1, rest 0.
    // 12 chained K=4 f32 WMMAs; uniform flow => EXEC all ones.
    v8f acc = {};
    const float sel0 = (m == 0) ? 1.f : 0.f;
    const float sel1 = (m == 1) ? 1.f : 0.f;
    #pragma unroll
    for (int ch = 0; ch < 12; ++ch) {
        int k0 = ch * 4 + khalf * 2;                       // even
        v2f a   = *(const v2f*)&ldsE[w * 16 + m][k0];      // A[m][k0], A[m][k0+1]
        v2f w0v = *(const v2f*)&ldsW[k0];
        v2f w1v = *(const v2f*)&ldsW[48 + k0];
        v2f bb  = w0v * sel0 + w1v * sel1;                 // B[k][n] (n=lane&15)
        acc = __builtin_amdgcn_wmma_f32_16x16x4_f32(
            false, a, false, bb, (short)0, acc, false, false);
    }

    // extract columns 0 (dot w0) and 1 (dot w1): lanes n in {0,1} (both halves)
    if (m < 2) {
        int base = m * 16 + khalf * 8;
        #pragma unroll
        for (int r = 0; r < 8; ++r) ldsDot[w][base + r] = acc[r];
    }
    ldsNum[tid] = (lane < 16) ? tok : 0.f;

    __syncthreads();

    // Per-timestep 2x2 transition matrix in the (logsumexp,+) semiring.
    if (lane < 16) {
        int row   = w * 16 + m;
        float dot0 = ldsDot[w][m];
        float dot1 = ldsDot[w][16 + m];
        float lp1  = ldsLp[row][0];
        float lp2  = ldsLp[row][1];
        float sO   = cw[96];
        float s0I  = cw[97];
        float s1I  = cw[98];
        float a00, a01, a10, a11;
        if (myT < lens[b]) {
            a00 = __logf(__expf(sO + lp1) + dot0);  // lse(sO+lp1, m0)
            a01 = __logf(dot1);                     // m1
            a10 = s0I + lp2;
            a11 = s1I + lp2;
        } else {                                    // masked step = identity
            a00 = 0.f; a11 = 0.f; a01 = NEGF; a10 = NEGF;
        }
        ldsM[row][0] = a00; ldsM[row][1] = a01;
        ldsM[row][2] = a10; ldsM[row][3] = a11;
    }
    __syncthreads();

    // In-order tree reduction of 128 matrices: result = M[127] ⊗ ... ⊗ M[0]
    for (int s = 1; s < 128; s <<= 1) {
        int i = tid * (s << 1);
        if (i + s < 128) {
            float b00 = ldsM[i][0],     b01 = ldsM[i][1],     b10 = ldsM[i][2],     b11 = ldsM[i][3];
            float a00 = ldsM[i + s][0], a01 = ldsM[i + s][1], a10 = ldsM[i + s][2], a11 = ldsM[i + s][3];
            float c00 = lse2(a00 + b00, a01 + b10);
            float c01 = lse2(a00 + b01, a01 + b11);
            float c10 = lse2(a10 + b00, a11 + b10);
            float c11 = lse2(a10 + b01, a11 + b11);
            ldsM[i][0] = c00; ldsM[i][1] = c01; ldsM[i][2] = c10; ldsM[i][3] = c11;
        }
        __syncthreads();
    }
    // deterministic numerator reduction
    for (int s = 128; s > 0; s >>= 1) {
        if (tid < s) ldsNum[tid] += ldsNum[tid + s];
        __syncthreads();
    }

    if (tid == 0) {
        float* sm = segMat + (size_t)(b * SEG + seg) * 4;
        sm[0] = ldsM[0][0]; sm[1] = ldsM[0][1]; sm[2] = ldsM[0][2]; sm[3] = ldsM[0][3];
        numPart[bid] = ldsNum[0];
    }
}

__global__ __launch_bounds__(128) void crf_final_kernel(
    const float* __restrict__ cw, const float* __restrict__ segMat,
    const float* __restrict__ numPart, float* __restrict__ out)
{
    __shared__ float dAcc[128];
    __shared__ float nAcc[128];
    const int tid = threadIdx.x;  // = batch index

    const float* sm = segMat + (size_t)tid * SEG * 4;
    float p00 = sm[0], p01 = sm[1], p10 = sm[2], p11 = sm[3];
    for (int s = 1; s < SEG; ++s) {
        const float* a = sm + s * 4;
        float a00 = a[0], a01 = a[1], a10 = a[2], a11 = a[3];
        float c00 = lse2(a00 + p00, a01 + p10);
        float c01 = lse2(a00 + p01, a01 + p11);
        float c10 = lse2(a10 + p00, a11 + p10);
        float c11 = lse2(a10 + p01, a11 + p11);
        p00 = c00; p01 = c01; p10 = c10; p11 = c11;
    }
    float alpha0 = lse2(p00, p01 + NEGF);  // alpha_T[0] with alpha_0 = [0, NEG]
    float den    = alpha0 + cw[99];        // + sfin

    float np = 0.f;
    for (int j = 0; j < 32; ++j) np += numPart[tid * 32 + j];

    dAcc[tid] = den;
    nAcc[tid] = np;
    __syncthreads();
    for (int s = 64; s > 0; s >>= 1) {
        if (tid < s) { dAcc[tid] += dAcc[tid + s]; nAcc[tid] += nAcc[tid + s]; }
        __syncthreads();
    }
    if (tid == 0) out[0] = nAcc[0] - dAcc[0];
}

extern "C" void kernel_launch(void* const* d_in, const int* in_sizes, int n_in,
                              void* d_out, int out_size, void* d_ws, size_t ws_size,
                              hipStream_t stream) {
    (void)in_sizes; (void)n_in; (void)out_size; (void)ws_size;
    const float* log_probs  = (const float*)d_in[0];
    const float* den_params = (const float*)d_in[1];
    const int*   input_lens = (const int*)d_in[2];
    const int*   labels     = (const int*)d_in[3];

    float* cw      = (float*)d_ws;
    float* segMat  = cw + WS_SEGMAT;
    float* numPart = cw + WS_NUMP;

    crf_consts_kernel<<<1, 32, 0, stream>>>(den_params, cw);
    crf_seg_kernel<<<BB * SEG, 256, 0, stream>>>(log_probs, input_lens, labels, cw,
                                                 segMat, numPart);
    crf_final_kernel<<<1, 128, 0, stream>>>(cw, segMat, numPart, (float*)d_out);
}